// DynamicCasHGNN_40716289966341
// MI455X (gfx1250) — compile-verified
//
#include <hip/hip_runtime.h>

// DynamicCasHGNN for MI455X (gfx1250, wave32).
// - GEMMs (conv transform + fuse MLP) use V_WMMA_F32_16X16X4_F32 tiles with
//   the 64x64 weight matrix staged in LDS (shared by all 8 waves/block) and
//   2 row-tiles register-blocked per wave (each LDS B-fragment feeds 2 WMMAs).
// - Hypergraph mean-aggregation uses global_atomic_add_f32 (L2-resident:
//   h=25.6MB, e=12.8MB, n=25.6MB all fit in 192MB L2).
// - Scan is done incrementally; carry lives in d_out (in-place fuse is safe:
//   each row is owned by exactly one wave, reads precede writes).

#define N_USERS 100000
#define E_HYPER 50000
#define NNZ_T   800000
#define T_STEPS 8
#define DIM     64
#define ROWT    2          // row-tiles (of 16 rows) per wave

typedef __attribute__((ext_vector_type(2))) float v2f;
typedef __attribute__((ext_vector_type(8))) float v8f;

// ---------------------------------------------------------------- zero fill
__global__ void zero_f32(float* __restrict__ p, long long n) {
  long long i = (long long)blockIdx.x * blockDim.x + threadIdx.x;
  const long long stride = (long long)gridDim.x * blockDim.x;
  for (; i < n; i += stride) p[i] = 0.0f;
}

// ------------------------------------------------- h = relu(A @ W + bias)
// Block: 8 waves x 32 rows = 256 rows. W staged in LDS once per block.
__global__ void gemm_relu_wmma(const float* __restrict__ A,
                               const float* __restrict__ W,
                               const float* __restrict__ bias,
                               float* __restrict__ H, int nrows) {
  __shared__ float Wl[DIM * DIM];
  for (int i = threadIdx.x; i < DIM * DIM / 4; i += blockDim.x)
    ((float4*)Wl)[i] = ((const float4*)W)[i];
  __syncthreads();

  const int lane = threadIdx.x & 31;
  const int wave = threadIdx.x >> 5;
  const int waveBase = (blockIdx.x * (blockDim.x >> 5) + wave) * (16 * ROWT);
  if (waveBase >= nrows) return;              // wave-uniform: EXEC stays full
  const int half = lane >> 4;                 // 0: K=0,1 / 1: K=2,3
  const int l16  = lane & 15;

  int arow0 = waveBase + l16;        if (arow0 >= nrows) arow0 = nrows - 1;
  int arow1 = waveBase + 16 + l16;   if (arow1 >= nrows) arow1 = nrows - 1;
  const float* __restrict__ ap0 = A + (size_t)arow0 * DIM;
  const float* __restrict__ ap1 = A + (size_t)arow1 * DIM;

  v8f c[ROWT][4];
#pragma unroll
  for (int rt = 0; rt < ROWT; ++rt)
#pragma unroll
    for (int nt = 0; nt < 4; ++nt) c[rt][nt] = (v8f){};

#pragma unroll
  for (int k = 0; k < DIM / 4; ++k) {
    const int ka = 4 * k + 2 * half;
    v2f a0; a0.x = ap0[ka]; a0.y = ap0[ka + 1];
    v2f a1; a1.x = ap1[ka]; a1.y = ap1[ka + 1];
#pragma unroll
    for (int nt = 0; nt < 4; ++nt) {
      v2f b;
      b.x = Wl[ka * DIM + 16 * nt + l16];
      b.y = Wl[(ka + 1) * DIM + 16 * nt + l16];
      c[0][nt] = __builtin_amdgcn_wmma_f32_16x16x4_f32(false, a0, false, b, (short)0, c[0][nt], false, false);
      c[1][nt] = __builtin_amdgcn_wmma_f32_16x16x4_f32(false, a1, false, b, (short)0, c[1][nt], false, false);
    }
  }

  float bv[4];
#pragma unroll
  for (int nt = 0; nt < 4; ++nt) bv[nt] = bias[16 * nt + l16];

#pragma unroll
  for (int rt = 0; rt < ROWT; ++rt) {
#pragma unroll
    for (int r = 0; r < 8; ++r) {
      const int row = waveBase + rt * 16 + r + half * 8;
      if (row < nrows) {
        float* __restrict__ op = H + (size_t)row * DIM;
#pragma unroll
        for (int nt = 0; nt < 4; ++nt)
          op[16 * nt + l16] = fmaxf(c[rt][nt][r] + bv[nt], 0.0f);
      }
    }
  }
}

// ------------------------------------ segment-sum scatter (+count) via atomics
// 64 consecutive threads (2 waves) handle one incidence: coalesced 128B
// reads of src row + coalesced atomic f32 adds into dst row (L2-resident).
__global__ void scatter_sum(const int* __restrict__ src_idx,
                            const int* __restrict__ dst_idx,
                            const float* __restrict__ src,
                            float* __restrict__ dst_sum,
                            float* __restrict__ dst_cnt) {
  const long long gid = (long long)blockIdx.x * blockDim.x + threadIdx.x;
  const int i = (int)(gid >> 6);
  const int d = (int)(gid & 63);
  if (i >= NNZ_T) return;
  const int si = src_idx[i];
  const int di = dst_idx[i];
  atomicAdd(&dst_sum[(size_t)di * DIM + d], src[(size_t)si * DIM + d]);
  if (d == 0) atomicAdd(&dst_cnt[di], 1.0f);
}

// ------------------------------------------------ row-wise mean normalization
__global__ void normalize_rows(float* __restrict__ buf,
                               const float* __restrict__ cnt, int rows) {
  const long long gid = (long long)blockIdx.x * blockDim.x + threadIdx.x;
  if (gid >= (long long)rows * DIM) return;
  const int r = (int)(gid >> 6);
  buf[gid] = buf[gid] / fmaxf(cnt[r], 1.0f);
}

// --------------------------------------------------------------- fuse step
// hidden = p0*hidden + p1*dy, p = softmax over the two candidate scores,
// score(x) = tanh(x @ w1 + b1) @ w2 + b2. The [32,64]x[64,64] parts are WMMA
// (w1 staged in LDS, 2 row-tiles/wave); the @w2 reduction is a shfl_xor
// butterfly within each 16-lane half (wave32-safe).
__global__ void fuse_wmma(float* __restrict__ hidden,        // in/out [N,D]
                          const float* __restrict__ dy,      // [N,D]
                          const float* __restrict__ w1,      // [D,D]
                          const float* __restrict__ b1,      // [D]
                          const float* __restrict__ w2,      // [D,1]
                          const float* __restrict__ b2,      // [1]
                          int nrows) {
  __shared__ float Wl[DIM * DIM];
  for (int i = threadIdx.x; i < DIM * DIM / 4; i += blockDim.x)
    ((float4*)Wl)[i] = ((const float4*)w1)[i];
  __syncthreads();

  const int lane = threadIdx.x & 31;
  const int wave = threadIdx.x >> 5;
  const int waveBase = (blockIdx.x * (blockDim.x >> 5) + wave) * (16 * ROWT);
  if (waveBase >= nrows) return;              // wave-uniform exit
  const int half = lane >> 4;
  const int l16  = lane & 15;

  int arow0 = waveBase + l16;        if (arow0 >= nrows) arow0 = nrows - 1;
  int arow1 = waveBase + 16 + l16;   if (arow1 >= nrows) arow1 = nrows - 1;

  float b1v[4], w2v[4];
#pragma unroll
  for (int nt = 0; nt < 4; ++nt) {
    b1v[nt] = b1[16 * nt + l16];
    w2v[nt] = w2[16 * nt + l16];
  }
  const float b2s = b2[0];

  float s[2][ROWT][8];
#pragma unroll
  for (int cand = 0; cand < 2; ++cand) {
    const float* __restrict__ src = (cand == 0) ? hidden : dy;
    const float* __restrict__ ap0 = src + (size_t)arow0 * DIM;
    const float* __restrict__ ap1 = src + (size_t)arow1 * DIM;

    v8f c[ROWT][4];
#pragma unroll
    for (int rt = 0; rt < ROWT; ++rt)
#pragma unroll
      for (int nt = 0; nt < 4; ++nt) c[rt][nt] = (v8f){};

#pragma unroll
    for (int k = 0; k < DIM / 4; ++k) {
      const int ka = 4 * k + 2 * half;
      v2f a0; a0.x = ap0[ka]; a0.y = ap0[ka + 1];
      v2f a1; a1.x = ap1[ka]; a1.y = ap1[ka + 1];
#pragma unroll
      for (int nt = 0; nt < 4; ++nt) {
        v2f b;
        b.x = Wl[ka * DIM + 16 * nt + l16];
        b.y = Wl[(ka + 1) * DIM + 16 * nt + l16];
        c[0][nt] = __builtin_amdgcn_wmma_f32_16x16x4_f32(false, a0, false, b, (short)0, c[0][nt], false, false);
        c[1][nt] = __builtin_amdgcn_wmma_f32_16x16x4_f32(false, a1, false, b, (short)0, c[1][nt], false, false);
      }
    }

#pragma unroll
    for (int rt = 0; rt < ROWT; ++rt) {
#pragma unroll
      for (int r = 0; r < 8; ++r) {
        float acc = tanhf(c[rt][0][r] + b1v[0]) * w2v[0]
                  + tanhf(c[rt][1][r] + b1v[1]) * w2v[1]
                  + tanhf(c[rt][2][r] + b1v[2]) * w2v[2]
                  + tanhf(c[rt][3][r] + b1v[3]) * w2v[3];
        // reduce over the 16 lanes of this half (xor offsets stay in-half)
        acc += __shfl_xor(acc, 1);
        acc += __shfl_xor(acc, 2);
        acc += __shfl_xor(acc, 4);
        acc += __shfl_xor(acc, 8);
        s[cand][rt][r] = acc + b2s;
      }
    }
  }

#pragma unroll
  for (int rt = 0; rt < ROWT; ++rt) {
#pragma unroll
    for (int r = 0; r < 8; ++r) {
      const int row = waveBase + rt * 16 + r + half * 8;
      const float s0 = s[0][rt][r], s1 = s[1][rt][r];
      const float m  = fmaxf(s0, s1);
      const float e0 = __expf(s0 - m), e1 = __expf(s1 - m);
      const float p0 = e0 / (e0 + e1);
      const float p1 = 1.0f - p0;
      if (row < nrows) {
        float* __restrict__ hp = hidden + (size_t)row * DIM;
        const float* __restrict__ dp = dy + (size_t)row * DIM;
#pragma unroll
        for (int nt = 0; nt < 4; ++nt) {
          const int col = 16 * nt + l16;
          hp[col] = p0 * hp[col] + p1 * dp[col];
        }
      }
    }
  }
}

// ------------------------------------------------------------------ launch
extern "C" void kernel_launch(void* const* d_in, const int* in_sizes, int n_in,
                              void* d_out, int out_size, void* d_ws, size_t ws_size,
                              hipStream_t stream) {
  const float* user_emb = (const float*)d_in[0];
  const float* W_conv   = (const float*)d_in[1];
  const float* b_conv   = (const float*)d_in[2];
  const float* fus_w1   = (const float*)d_in[3];
  const float* fus_b1   = (const float*)d_in[4];
  const float* fus_w2   = (const float*)d_in[5];
  const float* fus_b2   = (const float*)d_in[6];
  const int*   edge_nodes = (const int*)d_in[7];   // [T, NNZ]
  const int*   edge_hyper = (const int*)d_in[8];   // [T, NNZ]
  float* out = (float*)d_out;                       // carry / final [N, D]

  // Workspace layout (floats): ~64.6 MB total
  float* ws    = (float*)d_ws;
  float* h     = ws;                                   // N*D
  float* e_sum = h     + (size_t)N_USERS * DIM;        // E*D
  float* e_cnt = e_sum + (size_t)E_HYPER * DIM;        // E
  float* n_cnt = e_cnt + (size_t)E_HYPER;              // N
  float* dy    = n_cnt + (size_t)N_USERS;              // N*D

  const int rowsPerBlk = 8 * 16 * ROWT;                // 8 waves * 32 rows
  const int gemmBlocks = (N_USERS + rowsPerBlk - 1) / rowsPerBlk;
  const int scatBlocks = (int)(((long long)NNZ_T * DIM + 255) / 256);
  const int normEBlk   = (E_HYPER * DIM + 255) / 256;
  const int normNBlk   = (N_USERS * DIM + 255) / 256;

  // 1) h = relu(user_emb @ W_conv + b_conv)
  gemm_relu_wmma<<<gemmBlocks, 256, 0, stream>>>(user_emb, W_conv, b_conv, h, N_USERS);

  // 2) per-timestep hconv + incremental fuse scan
  for (int t = 0; t < T_STEPS; ++t) {
    const int* nd = edge_nodes + (size_t)t * NNZ_T;
    const int* hy = edge_hyper + (size_t)t * NNZ_T;
    float* target = (t == 0) ? out : dy;

    zero_f32<<<1024, 256, 0, stream>>>(e_sum, (long long)E_HYPER * DIM);
    zero_f32<<<64,   256, 0, stream>>>(e_cnt, (long long)E_HYPER);
    zero_f32<<<1024, 256, 0, stream>>>(target, (long long)N_USERS * DIM);
    zero_f32<<<128,  256, 0, stream>>>(n_cnt, (long long)N_USERS);

    // node -> hyperedge mean
    scatter_sum<<<scatBlocks, 256, 0, stream>>>(nd, hy, h, e_sum, e_cnt);
    normalize_rows<<<normEBlk, 256, 0, stream>>>(e_sum, e_cnt, E_HYPER);
    // hyperedge -> node mean
    scatter_sum<<<scatBlocks, 256, 0, stream>>>(hy, nd, e_sum, target, n_cnt);
    normalize_rows<<<normNBlk, 256, 0, stream>>>(target, n_cnt, N_USERS);

    if (t > 0) {
      fuse_wmma<<<gemmBlocks, 256, 0, stream>>>(out, dy, fus_w1, fus_b1,
                                                fus_w2, fus_b2, N_USERS);
    }
  }
}